// MaskedRead_60438779789437
// MI455X (gfx1250) — compile-verified
//
#include <hip/hip_runtime.h>
#include <hip/hip_bf16.h>

typedef float v2f __attribute__((ext_vector_type(2)));
typedef float v8f __attribute__((ext_vector_type(8)));

constexpr int NB = 4;      // batches
constexpr int DK = 128;    // key dim
constexpr int DV = 512;    // value dim
constexpr int NQ = 4096;   // query positions (TQ*H*W)
constexpr int NM = 8192;   // memory positions (TM*H*W)
constexpr float SCALE = 0.0883883476483184f; // 1/sqrt(128)
constexpr float NEGV  = -1.0e30f;

// ---------------------------------------------------------------------------
// Pre-pass: per-batch validity flag  valid[b] = qmask.any() && mmask.any()
// ---------------------------------------------------------------------------
__global__ __launch_bounds__(256) void mr_validity_kernel(
    const unsigned char* __restrict__ qmask,
    const unsigned char* __restrict__ mmask,
    int* __restrict__ valid)
{
    const int b = blockIdx.x;
    __shared__ int anyQ, anyM;
    if (threadIdx.x == 0) { anyQ = 0; anyM = 0; }
    __syncthreads();
    int lq = 0, lm = 0;
    for (int i = threadIdx.x; i < NQ; i += blockDim.x) lq |= qmask[b * NQ + i];
    for (int i = threadIdx.x; i < NM; i += blockDim.x) lm |= mmask[b * NM + i];
    if (lq) atomicOr(&anyQ, 1);
    if (lm) atomicOr(&anyM, 1);
    __syncthreads();
    if (threadIdx.x == 0) valid[b] = (anyQ && anyM) ? 1 : 0;
}

// ---------------------------------------------------------------------------
// Flash-attention main kernel.
// Grid: (NQ/16, NB). Block: 256 threads = 8 wave32.
// Each block handles 16 queries; each wave owns a 64-row slice of DV.
// Memory processed in blocks of 128 rows; wave w computes S for rows
// [mb + 16w, mb + 16w + 16) and the PV contraction for dv in [64w, 64w+64).
// fp32 WMMA (16x16x4) keeps full fp32 precision; problem is HBM/L2-bound.
//
// LDS layouts: Q and P tiles are stored pair-interleaved so a 4x16 f32
// B-fragment (vgpr0 <-> K=2*hi, vgpr1 <-> K=2*hi+1, col=laneLo) is a single
// aligned ds_load_b64 with conflict-free banking:
//   elem (k, q) at dword (k>>1)*32 + 2*q + (k&1)
//
// Both WMMA loops load operands in groups of 8 fragments (16 loads in
// flight) before issuing 8 back-to-back WMMAs, so the scheduler emits
// staggered s_wait_loadcnt/dscnt instead of wait-to-zero per WMMA.
// ---------------------------------------------------------------------------
__global__ __launch_bounds__(256) void mr_attn_kernel(
    const float* __restrict__ qkey,
    const float* __restrict__ qval,
    const unsigned char* __restrict__ qmask,
    const float* __restrict__ mkey,
    const float* __restrict__ mval,
    const unsigned char* __restrict__ mmask,
    float* __restrict__ out,
    const int* __restrict__ valid)
{
    const int b      = blockIdx.y;
    const int q0     = blockIdx.x * 16;
    const int tid    = threadIdx.x;
    const int w      = tid >> 5;        // wave id 0..7
    const int lane   = tid & 31;
    const int laneLo = lane & 15;
    const int hi     = lane >> 4;       // half-wave select

    __shared__ __align__(16) float sQ[DK * 16];   // Q tile, pair-interleaved
    __shared__ __align__(16) float sP[128 * 16];  // P block, pair-interleaved
    __shared__ float redmax[8 * 16];              // per-wave column maxima
    __shared__ float redsum[8 * 16];              // per-wave column sums
    __shared__ float sM[16];                      // running max per query
    __shared__ float sL[16];                      // running denominator per query

    if (tid < 16) { sM[tid] = -3.0e38f; sL[tid] = 0.0f; }

    // -- Stage Q tile [DK x 16] into LDS (pair-interleaved), coalesced loads.
    const float* qkb = qkey + (size_t)b * DK * NQ + q0;
#pragma unroll
    for (int i = 0; i < 8; ++i) {
        const int idx = tid + i * 256;       // 2048 elements total
        const int d   = idx >> 4;
        const int q   = idx & 15;
        sQ[((d >> 1) << 5) + 2 * q + (d & 1)] = qkb[(size_t)d * NQ + q];
    }

    v8f oacc[4];
#pragma unroll
    for (int t = 0; t < 4; ++t)
        oacc[t] = (v8f){0.f, 0.f, 0.f, 0.f, 0.f, 0.f, 0.f, 0.f};

    __syncthreads();

    const float*         mkb = mkey  + (size_t)b * DK * NM;
    const float*         mvb = mval  + (size_t)b * DV * NM;
    const unsigned char* mmb = mmask + (size_t)b * NM;

    for (int mb = 0; mb < NM; mb += 128) {
        const int m0  = mb + w * 16;
        const int m0c = m0 + laneLo;

        // Prefetch next K block for this wave (emits global_prefetch_b8).
        if (mb + 128 < NM)
            __builtin_prefetch(&mkb[(size_t)(2 * hi) * NM + m0c + 128], 0, 0);

        // ---- Phase A: S = K^T Q for this wave's 16 memory rows -----------
        v8f s = (v8f){0.f, 0.f, 0.f, 0.f, 0.f, 0.f, 0.f, 0.f};
#pragma unroll
        for (int g = 0; g < 4; ++g) {
            v2f av[8], bv[8];
#pragma unroll
            for (int j = 0; j < 8; ++j) {            // 16 loads in flight
                const int kk = g * 8 + j;
                const int d  = kk * 4 + 2 * hi;
                av[j].x = mkb[(size_t)d       * NM + m0c];
                av[j].y = mkb[(size_t)(d + 1) * NM + m0c];
                bv[j]   = *(const v2f*)&sQ[((2 * kk + hi) << 5) + 2 * laneLo];
            }
#pragma unroll
            for (int j = 0; j < 8; ++j)              // 8 back-to-back WMMAs
                s = __builtin_amdgcn_wmma_f32_16x16x4_f32(
                        false, av[j], false, bv[j], (short)0, s, false, false);
        }

        // mask + scale + per-column (query) max; element (r,lane): M=r+8*hi
        float cmax = -3.0e38f;
#pragma unroll
        for (int r = 0; r < 8; ++r) {
            const bool keep = mmb[m0 + r + 8 * hi] != 0;
            float v = s[r] * SCALE;
            v = keep ? v : NEGV;
            s[r] = v;
            cmax = fmaxf(cmax, v);
        }
        cmax = fmaxf(cmax, __shfl_xor(cmax, 16, 32));
        if (lane < 16) redmax[w * 16 + lane] = cmax;
        __syncthreads();

        // ---- Phase B: block max, online-softmax rescale factor -----------
        const float Mold = sM[laneLo];
        float bmax = redmax[laneLo];
#pragma unroll
        for (int ww = 1; ww < 8; ++ww) bmax = fmaxf(bmax, redmax[ww * 16 + laneLo]);
        const float Mnew  = fmaxf(Mold, bmax);
        const float alpha = __expf(Mold - Mnew);

        // ---- Phase C: P = exp(S - Mnew), stash in LDS, partial sums ------
        float psum = 0.f;
#pragma unroll
        for (int r = 0; r < 8; ++r) {
            const float p = __expf(s[r] - Mnew);
            psum += p;
            const int kg = w * 16 + r + 8 * hi;       // memory row in block
            sP[((kg >> 1) << 5) + 2 * laneLo + (r & 1)] = p;
        }
        psum += __shfl_xor(psum, 16, 32);
        if (lane < 16) redsum[w * 16 + lane] = psum;
        __syncthreads();

        // ---- Phase D: wave 0 updates running softmax state ---------------
        if (w == 0 && lane < 16) {
            float tot = 0.f;
#pragma unroll
            for (int ww = 0; ww < 8; ++ww) tot += redsum[ww * 16 + lane];
            sL[lane] = sL[lane] * alpha + tot;
            sM[lane] = Mnew;
        }

        // ---- Phase E: O = alpha*O + V_slice @ P --------------------------
#pragma unroll
        for (int t = 0; t < 4; ++t) {
            v8f o = oacc[t];
#pragma unroll
            for (int j = 0; j < 8; ++j) o[j] *= alpha;   // alpha per-query (lane)
            const float* vrow = mvb + (size_t)(w * 64 + t * 16 + laneLo) * NM + mb;
#pragma unroll
            for (int g = 0; g < 4; ++g) {
                v2f av[8], pv[8];
#pragma unroll
                for (int j = 0; j < 8; ++j) {        // 16 loads in flight
                    const int kk = g * 8 + j;
                    const int k  = kk * 4 + 2 * hi;
                    av[j] = *(const v2f*)(vrow + k);            // global_load_b64
                    pv[j] = *(const v2f*)&sP[((2 * kk + hi) << 5) + 2 * laneLo];
                }
#pragma unroll
                for (int j = 0; j < 8; ++j)          // 8 back-to-back WMMAs
                    o = __builtin_amdgcn_wmma_f32_16x16x4_f32(
                            false, av[j], false, pv[j], (short)0, o, false, false);
            }
            oacc[t] = o;
        }
        __syncthreads();   // protect sP / sM for next iteration
    }

    // ---- Epilogue: out = qval + (valid && qmask) * O / l -----------------
    const float l  = sL[laneLo];
    const bool  on = (valid[b] != 0) && (qmask[(size_t)b * NQ + q0 + laneLo] != 0);
    const float inv = on ? (1.0f / l) : 0.0f;

    float*       outb = out  + (size_t)b * DV * NQ + q0;
    const float* qvb  = qval + (size_t)b * DV * NQ + q0;
#pragma unroll
    for (int t = 0; t < 4; ++t) {
#pragma unroll
        for (int r = 0; r < 8; ++r) {
            const int dv = w * 64 + t * 16 + r + 8 * hi;
            outb[(size_t)dv * NQ + laneLo] =
                qvb[(size_t)dv * NQ + laneLo] + oacc[t][r] * inv;
        }
    }
}

// ---------------------------------------------------------------------------
extern "C" void kernel_launch(void* const* d_in, const int* in_sizes, int n_in,
                              void* d_out, int out_size, void* d_ws, size_t ws_size,
                              hipStream_t stream)
{
    const float*         qkey  = (const float*)d_in[0];
    const float*         qval  = (const float*)d_in[1];
    const unsigned char* qmask = (const unsigned char*)d_in[2];
    const float*         mkey  = (const float*)d_in[3];
    const float*         mval  = (const float*)d_in[4];
    const unsigned char* mmask = (const unsigned char*)d_in[5];
    float*               outp  = (float*)d_out;
    int*                 valid = (int*)d_ws;

    mr_validity_kernel<<<NB, 256, 0, stream>>>(qmask, mmask, valid);

    dim3 grid(NQ / 16, NB);
    mr_attn_kernel<<<grid, 256, 0, stream>>>(qkey, qval, qmask, mkey, mval, mmask,
                                             outp, valid);
}